// EdgeAugmentedConv_83949430768023
// MI455X (gfx1250) — compile-verified
//
#include <hip/hip_runtime.h>

// ---------------------------------------------------------------------------
// EdgeAugmentedConv for MI455X (gfx1250, wave32).
// Pipeline: zero -> xw=x@W (WMMA f32) -> edge MLP gate (WMMA f32 + LDS) ->
//           dis=rsqrt(deg) -> normalized atomic scatter -> LN+SiLU+residual.
// All GEMM work uses v_wmma_f32_16x16x4_f32 (native fp32 WMMA; the problem is
// bandwidth-bound, low-precision WMMA buys nothing here).
// ---------------------------------------------------------------------------

typedef float v2f __attribute__((ext_vector_type(2)));
typedef float v8f __attribute__((ext_vector_type(8)));

#define NN  100000
#define EE  1600000
#define CH  64
#define ED  16
#define HID 32

__device__ __forceinline__ float sigmoidf_(float v) {
    return 1.0f / (1.0f + __expf(-v));
}

// ---- K0: zero output accumulator and degree vector (every launch) ----------
__global__ void k_zero(float* __restrict__ out, float* __restrict__ deg) {
    int gid = blockIdx.x * blockDim.x + threadIdx.x;
    if (gid < NN * CH) out[gid] = 0.0f;
    if (gid < NN)      deg[gid] = 0.0f;
}

// ---- K1: xw = x @ W  (N x 64 @ 64 x 64), WMMA f32 16x16x4 ------------------
// One wave per 16x16 output tile. 8 waves/block: wave = (m_subtile, n_tile).
__global__ void __launch_bounds__(256) k_xw(const float* __restrict__ x,
                                            const float* __restrict__ W,
                                            float* __restrict__ xw) {
    const int lane   = threadIdx.x & 31;
    const int wave   = threadIdx.x >> 5;
    const int tile_n = wave & 3;                       // 4 col tiles (64/16)
    const int tile_m = blockIdx.x * 2 + (wave >> 2);   // 6250 row tiles
    const int m  = lane & 15;   // A row within tile / B,C col within tile
    const int hi = lane >> 4;   // half-wave selector
    const int kb = hi * 2;      // K sub-offset for A/B layout

    const float* xrow = x + (size_t)(tile_m * 16 + m) * CH;
    v8f c = {};
#pragma unroll
    for (int ks = 0; ks < 16; ++ks) {                  // K = 64 = 16 * 4
        v2f a, b;
        a.x = xrow[ks * 4 + kb];
        a.y = xrow[ks * 4 + kb + 1];
        b.x = W[(ks * 4 + kb)     * CH + tile_n * 16 + m];
        b.y = W[(ks * 4 + kb + 1) * CH + tile_n * 16 + m];
        c = __builtin_amdgcn_wmma_f32_16x16x4_f32(false, a, false, b,
                                                  (short)0, c, false, false);
    }
    float* orow = xw + (size_t)(tile_m * 16 + hi * 8) * CH + tile_n * 16 + m;
#pragma unroll
    for (int r = 0; r < 8; ++r) orow[(size_t)r * CH] = c[r];
}

// ---- K2: edge gate alpha = sigmoid(silu(ea@W1+b1)@W2+b2); deg += alpha -----
// One wave handles 32 edges: 2 A-tiles x 2 W1-col-tiles x 4 K-steps = 16 WMMA.
// Hidden activations staged in padded LDS, then one edge per lane for the dot.
__global__ void __launch_bounds__(256) k_edge(const float* __restrict__ ea,
                                              const long long* __restrict__ ei,
                                              const float* __restrict__ W1,
                                              const float* __restrict__ b1,
                                              const float* __restrict__ W2,
                                              const float* __restrict__ b2,
                                              float* __restrict__ alpha,
                                              float* deg) {
    __shared__ float lds[8][32][HID + 1];   // per-wave 32 edges x 32 hidden
    const int lane  = threadIdx.x & 31;
    const int wave  = threadIdx.x >> 5;
    const int wbase = (blockIdx.x * 8 + wave) * 32;
    const int m  = lane & 15;
    const int hi = lane >> 4;
    const int kb = hi * 2;

#pragma unroll
    for (int t = 0; t < 2; ++t) {                      // two 16-edge tiles
        const float* arow = ea + (size_t)(wbase + t * 16 + m) * ED;
#pragma unroll
        for (int nt = 0; nt < 2; ++nt) {               // HID = 32 = 2 * 16
            const float bias = b1[nt * 16 + m];        // bias is per-column
            v8f c = {bias, bias, bias, bias, bias, bias, bias, bias};
#pragma unroll
            for (int ks = 0; ks < 4; ++ks) {           // K = ED = 16 = 4 * 4
                v2f a, b;
                a.x = arow[ks * 4 + kb];
                a.y = arow[ks * 4 + kb + 1];
                b.x = W1[(ks * 4 + kb)     * HID + nt * 16 + m];
                b.y = W1[(ks * 4 + kb + 1) * HID + nt * 16 + m];
                c = __builtin_amdgcn_wmma_f32_16x16x4_f32(false, a, false, b,
                                                          (short)0, c, false, false);
            }
#pragma unroll
            for (int r = 0; r < 8; ++r) {
                float v = c[r];
                lds[wave][t * 16 + r + hi * 8][nt * 16 + m] = v * sigmoidf_(v);
            }
        }
    }
    __syncthreads();

    const int e = wbase + lane;          // one edge per lane
    float dot = b2[0];
#pragma unroll
    for (int j = 0; j < HID; ++j) dot += lds[wave][lane][j] * W2[j];
    const float al = sigmoidf_(dot);
    alpha[e] = al;
    const long long cl = ei[(size_t)EE + e];
    atomicAdd(deg + cl, al);
}

// ---- K3: dis = deg > 0 ? rsqrt(deg) : 0 (in place) -------------------------
__global__ void k_dis(float* deg) {
    int gid = blockIdx.x * blockDim.x + threadIdx.x;
    if (gid < NN) {
        float d = deg[gid];
        deg[gid] = d > 0.0f ? rsqrtf(d) : 0.0f;
    }
}

// ---- K4: out[col] += xw[row] * (dis[row]*alpha*dis[col]) -------------------
// 4 threads per edge, 16 channels each; xw and out are L2-resident (25.6 MB).
__global__ void __launch_bounds__(256) k_scatter(const long long* __restrict__ ei,
                                                 const float* __restrict__ alpha,
                                                 const float* __restrict__ dis,
                                                 const float* __restrict__ xw,
                                                 float* out) {
    const int gid = blockIdx.x * 256 + threadIdx.x;
    const int e = gid >> 2;
    const int q = gid & 3;
    const long long r  = ei[e];
    const long long cl = ei[(size_t)EE + e];
    const float nrm = alpha[e] * dis[r] * dis[cl];
    const float4* src = (const float4*)(xw + (size_t)r * CH + q * 16);
    float* dst = out + (size_t)cl * CH + q * 16;
#pragma unroll
    for (int i = 0; i < 4; ++i) {
        float4 v = src[i];
        atomicAdd(dst + i * 4 + 0, v.x * nrm);
        atomicAdd(dst + i * 4 + 1, v.y * nrm);
        atomicAdd(dst + i * 4 + 2, v.z * nrm);
        atomicAdd(dst + i * 4 + 3, v.w * nrm);
    }
}

// ---- K5: h = LN(acc + b)*gamma+beta; silu; + x (one wave per node) ---------
__global__ void __launch_bounds__(256) k_final(float* out,
                                               const float* __restrict__ x,
                                               const float* __restrict__ b,
                                               const float* __restrict__ gamma,
                                               const float* __restrict__ beta) {
    const int lane = threadIdx.x & 31;
    const int wave = threadIdx.x >> 5;
    const int n = blockIdx.x * 8 + wave;
    float h0 = out[(size_t)n * CH + lane]      + b[lane];
    float h1 = out[(size_t)n * CH + lane + 32] + b[lane + 32];

    float s = h0 + h1;
#pragma unroll
    for (int m = 16; m >= 1; m >>= 1) s += __shfl_xor(s, m, 32);
    const float mu = s * (1.0f / 64.0f);

    const float d0 = h0 - mu, d1 = h1 - mu;
    float v = d0 * d0 + d1 * d1;
#pragma unroll
    for (int m = 16; m >= 1; m >>= 1) v += __shfl_xor(v, m, 32);
    const float rs = rsqrtf(v * (1.0f / 64.0f) + 1e-5f);

    float y0 = d0 * rs * gamma[lane]      + beta[lane];
    float y1 = d1 * rs * gamma[lane + 32] + beta[lane + 32];
    y0 = y0 * sigmoidf_(y0) + x[(size_t)n * CH + lane];
    y1 = y1 * sigmoidf_(y1) + x[(size_t)n * CH + lane + 32];
    out[(size_t)n * CH + lane]      = y0;
    out[(size_t)n * CH + lane + 32] = y1;
}

// ---------------------------------------------------------------------------
extern "C" void kernel_launch(void* const* d_in, const int* in_sizes, int n_in,
                              void* d_out, int out_size, void* d_ws, size_t ws_size,
                              hipStream_t stream) {
    const float*     x     = (const float*)d_in[0];
    const long long* ei    = (const long long*)d_in[1];   // int64 edge_index [2,E]
    const float*     ea    = (const float*)d_in[2];
    const float*     W     = (const float*)d_in[3];
    const float*     b     = (const float*)d_in[4];
    const float*     W1    = (const float*)d_in[5];
    const float*     b1    = (const float*)d_in[6];
    const float*     W2    = (const float*)d_in[7];
    const float*     b2    = (const float*)d_in[8];
    const float*     gamma = (const float*)d_in[9];
    const float*     beta  = (const float*)d_in[10];
    float* out = (float*)d_out;

    // workspace layout: xw[N*64] | alpha[E] | deg->dis[N]  (~32.4 MB)
    float* xw    = (float*)d_ws;
    float* alpha = xw + (size_t)NN * CH;
    float* deg   = alpha + (size_t)EE;

    k_zero   <<<25000, 256, 0, stream>>>(out, deg);                  // 6.4M = NN*CH
    k_xw     <<<3125,  256, 0, stream>>>(x, W, xw);                  // 6250 m-tiles / 2
    k_edge   <<<6250,  256, 0, stream>>>(ea, ei, W1, b1, W2, b2, alpha, deg); // 256 edges/blk
    k_dis    <<<(NN + 255) / 256, 256, 0, stream>>>(deg);
    k_scatter<<<25000, 256, 0, stream>>>(ei, alpha, deg, xw, out);   // E*4 threads
    k_final  <<<12500, 256, 0, stream>>>(out, x, b, gamma, beta);    // 8 nodes/blk
}